// PairwiseModule_84920093376616
// MI455X (gfx1250) — compile-verified
//
#include <hip/hip_runtime.h>

typedef float v2f __attribute__((ext_vector_type(2)));
typedef float v4f __attribute__((ext_vector_type(4)));
typedef float v8f __attribute__((ext_vector_type(8)));

#define DIMK   256   // D
#define PAIR   128   // PAIR_DIM
#define WCOLS  512   // 2*D
#define BL     1024  // B*L
#define LSEQ   512   // L
#define NTOT   256   // 2*PAIR (p1 cols | p2 cols)

// ---------------------------------------------------------------------------
// Phase 1: P[m][n] = sum_k h[m][k] * Wcat[n][k]
//   Wcat[n][k] = (n < 128) ? W[n][k] : W[n-128][256+k]
// One wave32 per 16x16 tile of P; K swept 4 at a time with f32 WMMA.
// A 16x4 f32 layout: lanes 0-15 hold M=lane, v0=K0,v1=K1; lanes 16-31 K2,K3.
// B 4x16 f32 layout (mirror): lanes 0-15 hold N=lane, v0=K0,v1=K1; hi half K2,K3.
// C/D: VGPR r -> row (half*8 + r), col = lane & 15.
// ---------------------------------------------------------------------------
__global__ __launch_bounds__(256) void pairwise_gemm_f32_wmma(
    const float* __restrict__ h, const float* __restrict__ W,
    float* __restrict__ P)
{
    const int lane = threadIdx.x & 31;
    const int wave = (blockIdx.x * blockDim.x + threadIdx.x) >> 5; // 0..1023
    const int mt = wave >> 4;   // 0..63  (M tile)
    const int nt = wave & 15;   // 0..15  (N tile)
    const int m0 = mt << 4;
    const int n0 = nt << 4;

    const int half = lane >> 4;   // selects K pair {0,1} vs {2,3}
    const int sub  = lane & 15;   // row (A) / col (B) within tile

    // Per-lane A row: h[m0+sub][*]
    const float* __restrict__ arow = h + (m0 + sub) * DIMK;
    // Per-lane B "column" n mapped onto a W row (W1 vs W2 halves)
    const int n = n0 + sub;
    const float* __restrict__ brow =
        (n < PAIR) ? (W + n * WCOLS) : (W + (n - PAIR) * WCOLS + DIMK);

    v8f c = {0.f, 0.f, 0.f, 0.f, 0.f, 0.f, 0.f, 0.f};

    #pragma unroll 4
    for (int k0 = 0; k0 < DIMK; k0 += 4) {
        const int k = k0 + (half << 1);
        v2f a, b;
        a.x = arow[k];
        a.y = arow[k + 1];
        b.x = brow[k];
        b.y = brow[k + 1];
        // 8 args: (neg_a, A, neg_b, B, c_mod, C, reuse_a, reuse_b)
        c = __builtin_amdgcn_wmma_f32_16x16x4_f32(
                false, a, false, b, (short)0, c, false, false);
    }

    // Scatter C tile: VGPR r -> row m0 + half*8 + r, col n0 + sub
    float* __restrict__ prow = P + (m0 + (half << 3)) * NTOT + n0 + sub;
    #pragma unroll
    for (int r = 0; r < 8; ++r)
        prow[r * NTOT] = c[r];
}

// ---------------------------------------------------------------------------
// Phase 2: out[b,l,m,p] = p1[b,l,p] + p2[b,m,p] + bias[p]
// P[bl][0:128] = p1 row, P[bl][128:256] = p2 row.
// One block per (b,l). 256 threads = 8 m-subrows x 32 p-chunks(4 floats).
// 256 MB streamed out with non-temporal b128 stores (store-BW bound).
// ---------------------------------------------------------------------------
__global__ __launch_bounds__(256) void pairwise_bcast_add(
    const float* __restrict__ P, const float* __restrict__ bias,
    float* __restrict__ out)
{
    const int bl     = blockIdx.x;        // b*512 + l
    const int bidx   = bl >> 9;           // batch
    const int lane_p = threadIdx.x & 31;  // p chunk
    const int msub   = threadIdx.x >> 5;  // 0..7
    const int p4     = lane_p << 2;

    const v4f b4 = *(const v4f*)(bias + p4);
    const v4f p1 = *(const v4f*)(P + bl * NTOT + p4);
    const v4f s  = p1 + b4;               // hoisted p1 + bias

    const float* __restrict__ p2base = P + (bidx << 9) * NTOT + PAIR + p4;
    float* __restrict__ orow = out + (size_t)bl * LSEQ * PAIR + p4;

    #pragma unroll 4
    for (int m = msub; m < LSEQ; m += 8) {
        const v4f p2 = *(const v4f*)(p2base + m * NTOT);  // L2-resident
        const v4f v  = s + p2;
        __builtin_nontemporal_store(v, (v4f*)(orow + (size_t)m * PAIR));
    }
}

extern "C" void kernel_launch(void* const* d_in, const int* in_sizes, int n_in,
                              void* d_out, int out_size, void* d_ws, size_t ws_size,
                              hipStream_t stream) {
    const float* h    = (const float*)d_in[0];   // (2,512,256) f32
    const float* W    = (const float*)d_in[1];   // (128,512)   f32
    const float* bias = (const float*)d_in[2];   // (128,)      f32
    float* out = (float*)d_out;                  // (2,512,512,128) f32
    float* P   = (float*)d_ws;                   // 1024 x 256 f32 = 1 MB scratch

    // Phase 1: 1024 tiles, 1 wave each -> 128 blocks x 256 threads (8 waves)
    pairwise_gemm_f32_wmma<<<128, 256, 0, stream>>>(h, W, P);
    // Phase 2: one block per (b,l)
    pairwise_bcast_add<<<BL, 256, 0, stream>>>(P, bias, out);
}